// l1_l1_16819091931163
// MI455X (gfx1250) — compile-verified
//
#include <hip/hip_runtime.h>

// LISTA-style recurrence on MI455X (gfx1250), fp32 WMMA (V_WMMA_F32_16X16X4_F32).
//
//   * W is dead (first phi result always overwritten)
//   * cU = x_t @ P with P = A^T @ (A D / a)
//   * step:  h = phi(h_prev @ S + x_t @ P, h_prev @ G)
//   * out_t = h @ D^T
//
// v2: m/n-major LDS tiles (pitch 20 -> conflict-free ds_load_b64 fragments,
// no v_mov shuffles), double-buffered LDS with register-staged prefetch
// (1 barrier per K-chunk), 64x128 step tile (wave = 16x64, 8 accumulators,
// A-fragment feeds 8 WMMAs).

typedef __attribute__((ext_vector_type(2))) float v2f;
typedef __attribute__((ext_vector_type(8))) float v8f;

#define INPUT_N 256
#define COMP_N  64
#define HID_N   1024
#define BATCH_N 2048
#define T_N     64
#define LP      20   // LDS row pitch (floats): even + bank-spread

__device__ __forceinline__ v8f wmma_f32(v2f a, v2f b, v8f c) {
  return __builtin_amdgcn_wmma_f32_16x16x4_f32(false, a, false, b, (short)0, c,
                                               false, false);
}

// Reference phi: sequential masked writes (later conditions overwrite).
__device__ __forceinline__ float phi_f(float u, float v, float g1, float g2) {
  bool pos = (v >= 0.0f);
  bool neg = !pos;
  float t = 0.0f;
  if ((pos && (v + g1 + g2 <= u)) || (neg && (u >= g1 + g2)))
    t = u - g1 - g2;
  if ((pos && (v + g1 - g2 <= u) && (u < v + g1 + g2)) ||
      (neg && (v - g1 + g2 > u) && (u >= v - g1 - g2)))
    t = v;
  if ((pos && (g1 - g2 <= u) && (v + g1 - g2 > u)) ||
      (neg && (u < v - g1 - g2)))
    t = u - g1 + g2;
  if ((pos && (u >= -g1 - g2) && (u < g1 - g2)) ||
      (neg && (u >= -g1 + g2) && (u < g1 + g2)))
    t = 0.0f;
  if (pos && (u < -g1 - g2)) t = u + g1 + g2;
  if (neg && (u < v - g1 - g2)) t = u - g1 + g2;
  return t;
}

// ---------------------------------------------------------------------------
// Generic 64x64 fp32 WMMA GEMM, double-buffered. 256 thr = 8 waves (4Mx2N),
// wave tile 16x32 (2 accumulators).
// TA: A element (m,k) at Ap[k*lda+m];  TB: B element (k,n) at Bp[n*ldb+k]
// EPI 0: C=acc ; 1: C=acc, C2=acc/a ; 2: C = I - acc
// ---------------------------------------------------------------------------
template <bool TA, bool TB, int EPI>
__global__ __launch_bounds__(256) void gemm64(
    const float* __restrict__ Ap, int lda, const float* __restrict__ Bp,
    int ldb, float* __restrict__ Cp, float* __restrict__ Cp2, int ldc, int K,
    const float* __restrict__ a_ptr) {
  __shared__ float as[2][64][LP];
  __shared__ float bs[2][64][LP];

  const int bn = blockIdx.x << 6;
  const int bm = blockIdx.y << 6;
  const int tid = threadIdx.x;
  const int lane = tid & 31;
  const int wave = tid >> 5;
  const int wm = (wave & 3) << 4;
  const int wn = (wave >> 2) << 5;
  const int mrow = lane & 15;
  const int kh = (lane >> 4) << 1;

  // fill indices
  int a_i0, a_i1;
  if (!TA) { a_i0 = tid >> 2; a_i1 = (tid & 3) << 2; }
  else     { a_i0 = tid & 63; a_i1 = (tid >> 6) << 2; }
  const int b_i0 = tid & 63, b_i1 = (tid >> 6) << 2;

  v8f acc0 = {0.f, 0.f, 0.f, 0.f, 0.f, 0.f, 0.f, 0.f};
  v8f acc1 = acc0;

  float4 ra, rb;
  auto loadA = [&](int k0) {
    if (!TA) {
      ra = *(const float4*)(Ap + (size_t)(bm + a_i0) * lda + k0 + a_i1);
    } else {
      const float* p = Ap + (size_t)(k0 + a_i1) * lda + bm + a_i0;
      ra.x = p[0]; ra.y = p[lda]; ra.z = p[(size_t)2 * lda];
      ra.w = p[(size_t)3 * lda];
    }
  };
  auto loadB = [&](int k0) {
    if (TB) {
      rb = *(const float4*)(Bp + (size_t)(bn + b_i0) * ldb + k0 + b_i1);
    } else {
      const float* p = Bp + (size_t)(k0 + b_i1) * ldb + bn + b_i0;
      rb.x = p[0]; rb.y = p[ldb]; rb.z = p[(size_t)2 * ldb];
      rb.w = p[(size_t)3 * ldb];
    }
  };
  auto stage = [&](int buf) {
    *(float4*)&as[buf][a_i0][a_i1] = ra;
    *(float4*)&bs[buf][b_i0][b_i1] = rb;
  };

  loadA(0); loadB(0); stage(0);
  __syncthreads();

  const int NC = K >> 4;
  for (int c = 0; c < NC; ++c) {
    const int cur = c & 1;
    if (c + 1 < NC) { loadA((c + 1) << 4); loadB((c + 1) << 4); }
#pragma unroll
    for (int kk = 0; kk < 16; kk += 4) {
      v2f a  = *(const v2f*)&as[cur][wm + mrow][kk + kh];
      v2f b0 = *(const v2f*)&bs[cur][wn + mrow][kk + kh];
      v2f b1 = *(const v2f*)&bs[cur][wn + 16 + mrow][kk + kh];
      acc0 = wmma_f32(a, b0, acc0);
      acc1 = wmma_f32(a, b1, acc1);
    }
    if (c + 1 < NC) stage(cur ^ 1);
    __syncthreads();
  }

  const int rbase = bm + wm + ((lane >> 4) << 3);
  const int c0 = bn + wn + mrow;
  const float scale = (EPI == 1) ? (1.0f / a_ptr[0]) : 0.0f;
#pragma unroll
  for (int r = 0; r < 8; ++r) {
    const int row = rbase + r;
    float u0 = acc0[r], u1 = acc1[r];
    if (EPI == 2) {
      u0 = ((row == c0) ? 1.0f : 0.0f) - u0;
      u1 = ((row == c0 + 16) ? 1.0f : 0.0f) - u1;
    }
    Cp[(size_t)row * ldc + c0] = u0;
    Cp[(size_t)row * ldc + c0 + 16] = u1;
    if (EPI == 1) {
      Cp2[(size_t)row * ldc + c0] = u0 * scale;
      Cp2[(size_t)row * ldc + c0 + 16] = u1 * scale;
    }
  }
}

// ---------------------------------------------------------------------------
// Fused timestep:  Hnew = phi( Hprev@S + Xt@P , Hprev@G )
// Block tile 64(M) x 128(N); 8 waves (4Mx2N), wave tile 16x64.
// Loop 1 (K=HID): H A-tile feeds both S and G (8 WMMA per A-fragment).
// Loop 2 (K=INPUT): Xt@P into u accumulators only. phi in registers.
// ---------------------------------------------------------------------------
__global__ __launch_bounds__(256) void step_kernel(
    const float* __restrict__ Hprev, const float* __restrict__ S,
    const float* __restrict__ G, const float* __restrict__ Xt,
    const float* __restrict__ P, float* __restrict__ Hnew,
    const float* __restrict__ l1p, const float* __restrict__ l2p,
    const float* __restrict__ ap) {
  __shared__ float as[2][64][LP];
  __shared__ float sb[2][128][LP];
  __shared__ float gb[2][128][LP];

  const int bn = blockIdx.x << 7;
  const int bm = blockIdx.y << 6;
  const int tid = threadIdx.x;
  const int lane = tid & 31;
  const int wave = tid >> 5;
  const int wm = (wave & 3) << 4;
  const int wn = (wave >> 2) << 6;  // 0 or 64
  const int mrow = lane & 15;
  const int kh = (lane >> 4) << 1;

  const int a_i0 = tid >> 2, a_i1 = (tid & 3) << 2;
  const int b_i0 = tid & 127, b_i1 = (tid >> 7) << 3;  // 0 or 8

  v8f zero = {0.f, 0.f, 0.f, 0.f, 0.f, 0.f, 0.f, 0.f};
  v8f u[4] = {zero, zero, zero, zero};
  v8f v[4] = {zero, zero, zero, zero};

  float4 ra, rs0, rs1, rg0, rg1;
  auto loadA = [&](const float* M, int lda, int k0) {
    ra = *(const float4*)(M + (size_t)(bm + a_i0) * lda + k0 + a_i1);
  };
  auto loadBB = [&](const float* M, int k0, float4& q0, float4& q1) {
    const float* p = M + (size_t)(k0 + b_i1) * HID_N + bn + b_i0;
    q0.x = p[0];
    q0.y = p[HID_N];
    q0.z = p[2 * HID_N];
    q0.w = p[3 * HID_N];
    q1.x = p[4 * HID_N];
    q1.y = p[5 * HID_N];
    q1.z = p[6 * HID_N];
    q1.w = p[7 * HID_N];
  };

  // ---- loop 1: K = HID over Hprev with S (->u) and G (->v) ----
  loadA(Hprev, HID_N, 0);
  loadBB(S, 0, rs0, rs1);
  loadBB(G, 0, rg0, rg1);
  *(float4*)&as[0][a_i0][a_i1] = ra;
  *(float4*)&sb[0][b_i0][b_i1] = rs0;
  *(float4*)&sb[0][b_i0][b_i1 + 4] = rs1;
  *(float4*)&gb[0][b_i0][b_i1] = rg0;
  *(float4*)&gb[0][b_i0][b_i1 + 4] = rg1;
  __syncthreads();

  const int NC1 = HID_N >> 4;
  for (int c = 0; c < NC1; ++c) {
    const int cur = c & 1;
    if (c + 1 < NC1) {
      loadA(Hprev, HID_N, (c + 1) << 4);
      loadBB(S, (c + 1) << 4, rs0, rs1);
      loadBB(G, (c + 1) << 4, rg0, rg1);
    }
#pragma unroll
    for (int kk = 0; kk < 16; kk += 4) {
      v2f a = *(const v2f*)&as[cur][wm + mrow][kk + kh];
#pragma unroll
      for (int s = 0; s < 4; ++s) {
        v2f bS = *(const v2f*)&sb[cur][wn + (s << 4) + mrow][kk + kh];
        u[s] = wmma_f32(a, bS, u[s]);
        v2f bG = *(const v2f*)&gb[cur][wn + (s << 4) + mrow][kk + kh];
        v[s] = wmma_f32(a, bG, v[s]);
      }
    }
    if (c + 1 < NC1) {
      const int nxt = cur ^ 1;
      *(float4*)&as[nxt][a_i0][a_i1] = ra;
      *(float4*)&sb[nxt][b_i0][b_i1] = rs0;
      *(float4*)&sb[nxt][b_i0][b_i1 + 4] = rs1;
      *(float4*)&gb[nxt][b_i0][b_i1] = rg0;
      *(float4*)&gb[nxt][b_i0][b_i1 + 4] = rg1;
    }
    __syncthreads();
  }

  // ---- loop 2: K = INPUT over Xt and P (->u only) ----
  loadA(Xt, INPUT_N, 0);
  loadBB(P, 0, rs0, rs1);
  *(float4*)&as[0][a_i0][a_i1] = ra;
  *(float4*)&sb[0][b_i0][b_i1] = rs0;
  *(float4*)&sb[0][b_i0][b_i1 + 4] = rs1;
  __syncthreads();

  const int NC2 = INPUT_N >> 4;
  for (int c = 0; c < NC2; ++c) {
    const int cur = c & 1;
    if (c + 1 < NC2) {
      loadA(Xt, INPUT_N, (c + 1) << 4);
      loadBB(P, (c + 1) << 4, rs0, rs1);
    }
#pragma unroll
    for (int kk = 0; kk < 16; kk += 4) {
      v2f a = *(const v2f*)&as[cur][wm + mrow][kk + kh];
#pragma unroll
      for (int s = 0; s < 4; ++s) {
        v2f bS = *(const v2f*)&sb[cur][wn + (s << 4) + mrow][kk + kh];
        u[s] = wmma_f32(a, bS, u[s]);
      }
    }
    if (c + 1 < NC2) {
      const int nxt = cur ^ 1;
      *(float4*)&as[nxt][a_i0][a_i1] = ra;
      *(float4*)&sb[nxt][b_i0][b_i1] = rs0;
      *(float4*)&sb[nxt][b_i0][b_i1 + 4] = rs1;
    }
    __syncthreads();
  }

  // ---- phi epilogue ----
  const float a_s = ap[0];
  const float g1 = l1p[0] / a_s;
  const float g2 = l2p[0] / a_s;
  const int rbase = bm + wm + ((lane >> 4) << 3);
  const int c0 = bn + wn + mrow;
#pragma unroll
  for (int s = 0; s < 4; ++s) {
#pragma unroll
    for (int r = 0; r < 8; ++r) {
      Hnew[(size_t)(rbase + r) * HID_N + c0 + (s << 4)] =
          phi_f(u[s][r], v[s][r], g1, g2);
    }
  }
}

// ---------------------------------------------------------------------------
extern "C" void kernel_launch(void* const* d_in, const int* in_sizes, int n_in,
                              void* d_out, int out_size, void* d_ws,
                              size_t ws_size, hipStream_t stream) {
  (void)in_sizes; (void)n_in; (void)out_size; (void)ws_size;
  const float* data = (const float*)d_in[0];  // (T, B, INPUT)
  const float* A    = (const float*)d_in[1];  // (COMP, INPUT)
  const float* D    = (const float*)d_in[2];  // (INPUT, HID)
  const float* G    = (const float*)d_in[3];  // (HID, HID)
  const float* h0   = (const float*)d_in[4];  // (B, HID)
  const float* l1   = (const float*)d_in[5];
  const float* l2   = (const float*)d_in[6];
  const float* a    = (const float*)d_in[7];
  float* out = (float*)d_out;                 // (T, B, INPUT)

  float* ws  = (float*)d_ws;
  float* AD  = ws; ws += (size_t)COMP_N * HID_N;   // A@D
  float* ADs = ws; ws += (size_t)COMP_N * HID_N;   // (A@D)/a
  float* P   = ws; ws += (size_t)INPUT_N * HID_N;  // A^T @ ADs
  float* S   = ws; ws += (size_t)HID_N * HID_N;    // I - ADs^T @ AD
  float* H0  = ws; ws += (size_t)BATCH_N * HID_N;  // ping
  float* H1  = ws; ws += (size_t)BATCH_N * HID_N;  // pong

  // AD = A @ D ; ADs = AD / a
  gemm64<false, false, 1><<<dim3(HID_N / 64, COMP_N / 64), 256, 0, stream>>>(
      A, INPUT_N, D, HID_N, AD, ADs, HID_N, INPUT_N, a);
  // P = A^T @ ADs
  gemm64<true, false, 0><<<dim3(HID_N / 64, INPUT_N / 64), 256, 0, stream>>>(
      A, INPUT_N, ADs, HID_N, P, nullptr, HID_N, COMP_N, nullptr);
  // S = I - ADs^T @ AD
  gemm64<true, false, 2><<<dim3(HID_N / 64, HID_N / 64), 256, 0, stream>>>(
      ADs, HID_N, AD, HID_N, S, nullptr, HID_N, COMP_N, nullptr);

  for (int t = 0; t < T_N; ++t) {
    const float* hp = (t == 0) ? h0 : ((t & 1) ? H0 : H1);
    float* hn = (t & 1) ? H1 : H0;
    step_kernel<<<dim3(HID_N / 128, BATCH_N / 64), 256, 0, stream>>>(
        hp, S, G, data + (size_t)t * BATCH_N * INPUT_N, P, hn, l1, l2, a);
    // out_t = Hnew @ D^T
    gemm64<false, true, 0><<<dim3(INPUT_N / 64, BATCH_N / 64), 256, 0,
                             stream>>>(hn, HID_N, D, HID_N,
                                       out + (size_t)t * BATCH_N * INPUT_N,
                                       nullptr, INPUT_N, HID_N, nullptr);
  }
}